// ClipFastRCNNOutputLayers_24807731101777
// MI455X (gfx1250) — compile-verified
//
#include <hip/hip_runtime.h>
#include <stdint.h>

// ---------------- problem constants ----------------
#define RN 65536
#define KC 80
#define KP1 81
#define NTOT (RN * KC)          // 5,242,880 scores
#define WIMG 1333.0f
#define HIMG 800.0f
#define SCORE_TH 0.05f
#define NMS_TH 0.5f
#define MSEL 2048
#define NTOP 100
#define NB 8192                 // histogram bins over (0.05, 1.0]
#define CAP 4096                // candidate capacity (>= MSEL + max bin count)
#define RECF 8                  // floats per candidate record (32 B AoS)

// ---------------- workspace layout (bytes) ----------------
#define O_HIST   0u             // 8192 u32   = 32768
#define O_CNT    32768u         // 1 u32
#define O_THR    32776u         // 1 f32
#define O_VALID  32800u         // 64 u32
#define O_KEEP   33056u         // 64 u32
#define O_KEYS   33312u         // 4096 u64   = 32768  (8B aligned)
#define O_REC    66080u         // 2048*8 f32 = 65536  (32B records)
#define O_SUP    131616u        // 2048*64 u32 = 524288 -> total ~641 KB

typedef float v2f __attribute__((ext_vector_type(2)));
typedef float v8f __attribute__((ext_vector_type(8)));
typedef unsigned tdm_v4u __attribute__((ext_vector_type(4)));
typedef int tdm_v8i __attribute__((ext_vector_type(8)));
typedef int tdm_v4i __attribute__((ext_vector_type(4)));

// ---------------- TDM: 1-D tile DMA  global -> LDS  (ISA §8 D# layout) -------
// data_size = 4 B elements; tile = nelem elements; tensor_dim0 = tensor_len.
#if __has_builtin(__builtin_amdgcn_tensor_load_to_lds)
#define HAVE_TDM 1
__device__ __forceinline__ void tdm_load_1d(const void* gsrc, unsigned lds_off,
                                            unsigned nelem, unsigned tensor_len) {
  unsigned long long ga = (unsigned long long)(uintptr_t)gsrc;
  tdm_v4u g0;
  g0[0] = 1u;                                                // count=1, user D#
  g0[1] = lds_off;                                           // LDS byte address
  g0[2] = (unsigned)ga;                                      // global_addr[31:0]
  g0[3] = (unsigned)((ga >> 32) & 0x1FFFFFFu) | (2u << 30);  // addr[56:32] | type=2
  tdm_v8i g1;
  g1[0] = (int)(2u << 16);                                   // data_size = 4 B
  g1[1] = (int)((tensor_len & 0xFFFFu) << 16);               // tensor_dim0[15:0]
  g1[2] = (int)(((tensor_len >> 16) & 0xFFFFu) | (1u << 16));// dim0[31:16] | tensor_dim1=1
  g1[3] = (int)(nelem << 16);                                // tile_dim0 = nelem
  g1[4] = 1;                                                 // tile_dim1=1, tile_dim2=0
  g1[5] = (int)tensor_len;                                   // dim0_stride lo
  g1[6] = 0;                                                 // dim0_stride hi, dim1_stride lo
  g1[7] = 0;
  tdm_v4i gz = {0, 0, 0, 0};
#if defined(__clang_major__) && __clang_major__ >= 23
  tdm_v8i gz8 = {0, 0, 0, 0, 0, 0, 0, 0};
  __builtin_amdgcn_tensor_load_to_lds(g0, g1, gz, gz, gz8, 0);
#else
  __builtin_amdgcn_tensor_load_to_lds(g0, g1, gz, gz, 0);
#endif
}
#else
#define HAVE_TDM 0
#endif

// ---------------- phase 0: init counters ----------------
__global__ void k_init(unsigned* ws) {
  int i = blockIdx.x * blockDim.x + threadIdx.x;
  if (i < NB) ws[O_HIST / 4 + i] = 0u;
  if (i == 0) ws[O_CNT / 4] = 0u;
  if (i < 64) { ws[O_VALID / 4 + i] = 0u; ws[O_KEEP / 4 + i] = 0u; }
}

// ---------------- phase 1a: score histogram (LDS-local, then merge) ----------
__global__ void k_hist(const float* __restrict__ scores, unsigned* __restrict__ hist) {
  __shared__ unsigned lh[NB];
  for (int t = threadIdx.x; t < NB; t += blockDim.x) lh[t] = 0u;
  __syncthreads();
  long long stride = (long long)gridDim.x * blockDim.x;
  for (long long i = (long long)blockIdx.x * blockDim.x + threadIdx.x; i < NTOT; i += stride) {
    int r = (int)(i / KC);
    int c = (int)(i - (long long)r * KC);
    float s = scores[(long long)r * KP1 + c];
    if (s > SCORE_TH) {
      int b = (int)((s - SCORE_TH) * ((float)NB / (1.0f - SCORE_TH)));
      b = b < 0 ? 0 : (b > NB - 1 ? NB - 1 : b);
      atomicAdd(&lh[b], 1u);
    }
  }
  __syncthreads();
  for (int t = threadIdx.x; t < NB; t += blockDim.x)
    if (lh[t]) atomicAdd(&hist[t], lh[t]);
}

// ---------------- phase 1b: find threshold so that count(>thr) >= MSEL -------
__global__ void k_thresh(const unsigned* __restrict__ hist, float* thr) {
  if (blockIdx.x == 0 && threadIdx.x == 0) {
    unsigned cum = 0u;
    float t = SCORE_TH;
    for (int b = NB - 1; b >= 0; --b) {
      cum += hist[b];
      if (cum >= MSEL) {
        t = SCORE_TH + (float)b * ((1.0f - SCORE_TH) / (float)NB);
        break;
      }
    }
    *thr = t;
  }
}

// ---------------- phase 1c: collect candidates above threshold ---------------
__global__ void k_collect(const float* __restrict__ scores, const float* __restrict__ thr,
                          unsigned* __restrict__ cnt, unsigned long long* __restrict__ keys) {
  float t = *thr;
  long long stride = (long long)gridDim.x * blockDim.x;
  for (long long i = (long long)blockIdx.x * blockDim.x + threadIdx.x; i < NTOT; i += stride) {
    int r = (int)(i / KC);
    int c = (int)(i - (long long)r * KC);
    float s = scores[(long long)r * KP1 + c];
    if (s > t) {
      unsigned p = atomicAdd(cnt, 1u);
      if (p < CAP) {
        // key: score bits high (positive floats are monotone as u32),
        // inverted flat index low -> descending sort == top_k tie-break
        unsigned sb = __float_as_uint(s);
        keys[p] = ((unsigned long long)sb << 32) |
                  (unsigned long long)(0xFFFFFFFFu - (unsigned)i);
      }
    }
  }
}

// ---------------- phase 1d: single-block bitonic sort + gather/clip ----------
// Emits 32-byte AoS records {x1,y1,x2,y2,area,cls,score,0} so the IoU phase
// streams two contiguous slabs per tile through the TDM.
__global__ void k_sortgather(const float* __restrict__ boxes, const unsigned* __restrict__ cntp,
                             const unsigned long long* __restrict__ keys,
                             float* __restrict__ rec, unsigned* __restrict__ validw) {
  __shared__ unsigned long long sk[CAP];     // 32 KB LDS
  int tid = threadIdx.x;
  unsigned cnt = *cntp;
  if (cnt > CAP) cnt = CAP;
  for (int t = tid; t < CAP; t += blockDim.x) sk[t] = (t < (int)cnt) ? keys[t] : 0ULL;
  for (int k = 2; k <= CAP; k <<= 1) {
    for (int j = k >> 1; j > 0; j >>= 1) {
      __syncthreads();
      for (int t = tid; t < CAP; t += blockDim.x) {
        int ixj = t ^ j;
        if (ixj > t) {
          unsigned long long A = sk[t], B = sk[ixj];
          bool descBlk = ((t & k) == 0);       // descending overall
          if ((A < B) == descBlk) { sk[t] = B; sk[ixj] = A; }
        }
      }
    }
  }
  __syncthreads();
  for (int t = tid; t < MSEL; t += blockDim.x) {
    unsigned long long key = sk[t];
    float s = __uint_as_float((unsigned)(key >> 32));
    unsigned flat = 0xFFFFFFFFu - (unsigned)(key & 0xFFFFFFFFull);
    bool valid = (key != 0ULL);
    float x1 = 0.f, y1 = 0.f, x2 = 0.f, y2 = 0.f, ar = 0.f, cls = -1.f, sv = 0.f;
    if (valid) {
      unsigned r = flat / KC;
      unsigned c = flat - r * KC;
      const float* bp = boxes + (size_t)r * (KC * 4) + (size_t)c * 4;
      x1 = fminf(fmaxf(bp[0], 0.f), WIMG);
      y1 = fminf(fmaxf(bp[1], 0.f), HIMG);
      x2 = fminf(fmaxf(bp[2], 0.f), WIMG);
      y2 = fminf(fmaxf(bp[3], 0.f), HIMG);
      ar = fmaxf(x2 - x1, 0.f) * fmaxf(y2 - y1, 0.f);
      cls = (float)c;
      sv = s;
      atomicOr(&validw[t >> 5], 1u << (t & 31));
    }
    float* rp = rec + (size_t)t * RECF;
    rp[0] = x1; rp[1] = y1; rp[2] = x2; rp[3] = y2;
    rp[4] = ar; rp[5] = cls; rp[6] = sv; rp[7] = 0.f;
  }
}

// ---------------- phase 2: suppression bitmask via TDM + WMMA ----------------
// One wave per 16-row x 32-col tile. TDM DMAs the row slab (512 B) and col
// slab (1 KB) of candidate records into LDS; the union base area_i + area_j
// for each 16x16 sub-tile is the rank-2 outer product computed exactly by
// V_WMMA_F32_16X16X4_F32: A[:,0]=area_row, A[:,1]=1 ; B[0,:]=1, B[1,:]=area_col.
__global__ void __launch_bounds__(32) k_sup(const float* __restrict__ rec,
                                            unsigned* __restrict__ sup) {
  __shared__ float lrec[(16 + 32) * RECF];   // rows [0..127], cols [128..383]
  int lane = threadIdx.x;
  int blk = blockIdx.x;
  int tjw = blk & 63;                 // 32-column word index
  int rowbase = (blk >> 6) << 4;      // 16-row tile base
  int col0 = tjw * 32;

#if HAVE_TDM
  // LDS aperture addresses truncate to the wave-relative LDS byte offset.
  unsigned lbase = (unsigned)(uintptr_t)(const void*)&lrec[0];
  tdm_load_1d(rec + (size_t)rowbase * RECF, lbase, 16 * RECF, MSEL * RECF);
  tdm_load_1d(rec + (size_t)col0 * RECF, lbase + 16u * RECF * 4u, 32 * RECF, MSEL * RECF);
  __builtin_amdgcn_s_wait_tensorcnt(0);
  __asm__ __volatile__("" ::: "memory");   // keep LDS reads below the wait
#else
  for (int t = lane; t < 16 * RECF; t += 32) lrec[t] = rec[(size_t)rowbase * RECF + t];
  for (int t = lane; t < 32 * RECF; t += 32)
    lrec[16 * RECF + t] = rec[(size_t)col0 * RECF + t];
  __syncthreads();
#endif

  int half = lane >> 4;
  int nloc = lane & 15;
  bool lo = lane < 16;
  const float* lr = &lrec[0];
  const float* lc = &lrec[16 * RECF];

  // A: 16x4 f32, lanes 0-15 hold M=lane; upper lanes (K=2,3 slots) zeroed.
  v2f a;
  a.x = lo ? lr[lane * RECF + 4] : 0.f;
  a.y = lo ? 1.f : 0.f;
  v2f b0, b1;
  b0.x = lo ? 1.f : 0.f;
  b0.y = lo ? lc[lane * RECF + 4] : 0.f;
  b1.x = b0.x;
  b1.y = lo ? lc[(16 + lane) * RECF + 4] : 0.f;
  v8f cz = {};
  // D[m][n] = area_row[m] + area_col[n]  (exact f32 FMA)
  v8f d0 = __builtin_amdgcn_wmma_f32_16x16x4_f32(false, a, false, b0, (short)0, cz, false, false);
  v8f d1 = __builtin_amdgcn_wmma_f32_16x16x4_f32(false, a, false, b1, (short)0, cz, false, false);

  const float* c0 = &lc[nloc * RECF];
  const float* c1 = &lc[(16 + nloc) * RECF];
  float c0x1 = c0[0], c0y1 = c0[1], c0x2 = c0[2], c0y2 = c0[3], cls0 = c0[5];
  float c1x1 = c1[0], c1y1 = c1[1], c1x2 = c1[2], c1y2 = c1[3], cls1 = c1[5];

#pragma unroll
  for (int v = 0; v < 8; ++v) {
    const float* rp = &lr[(v + 8 * half) * RECF];  // C/D layout: lanes16-31 -> M=v+8
    float rx1 = rp[0], ry1 = rp[1], rx2 = rp[2], ry2 = rp[3], rc = rp[5];

    float iw0 = fmaxf(fminf(rx2, c0x2) - fmaxf(rx1, c0x1), 0.f);
    float ih0 = fmaxf(fminf(ry2, c0y2) - fmaxf(ry1, c0y1), 0.f);
    float in0 = iw0 * ih0;
    float un0 = d0[v] - in0;
    bool p0 = (rc == cls0) && (un0 > 0.f) && (in0 > NMS_TH * un0);

    float iw1 = fmaxf(fminf(rx2, c1x2) - fmaxf(rx1, c1x1), 0.f);
    float ih1 = fmaxf(fminf(ry2, c1y2) - fmaxf(ry1, c1y1), 0.f);
    float in1 = iw1 * ih1;
    float un1 = d1[v] - in1;
    bool p1 = (rc == cls1) && (un1 > 0.f) && (in1 > NMS_TH * un1);

    unsigned m0 = __builtin_amdgcn_ballot_w32(p0);   // rows v / v+8, cols 0-15
    unsigned m1 = __builtin_amdgcn_ballot_w32(p1);   // rows v / v+8, cols 16-31
    if (lane == 0) {
      sup[(size_t)(rowbase + v) * 64 + tjw]     = (m0 & 0xFFFFu) | (m1 << 16);
      sup[(size_t)(rowbase + v + 8) * 64 + tjw] = (m0 >> 16) | (m1 & 0xFFFF0000u);
    }
  }
}

// ---------------- phase 3: sequential greedy suppression scan ----------------
__global__ void k_nms(const unsigned* __restrict__ sup, const unsigned* __restrict__ validw,
                      unsigned* __restrict__ keepw) {
  __shared__ unsigned removed[64];
  int t = threadIdx.x;                // 64 threads, one 32-bit word each
  removed[t] = ~validw[t];
  __syncthreads();
  for (int i = 0; i < MSEL; ++i) {
    bool alive = ((removed[i >> 5] >> (i & 31)) & 1u) == 0u;
    if (alive) {
      unsigned w = sup[(size_t)i * 64 + t];
      int iw = i >> 5;
      if (t < iw) w = 0u;
      else if (t == iw) {
        int sh = i & 31;
        w = (sh == 31) ? 0u : (w & (0xFFFFFFFFu << (sh + 1)));   // only j > i
      }
      removed[t] |= w;
    }
    __syncthreads();
  }
  keepw[t] = ~removed[t];
}

// ---------------- phase 4: emit top-100 kept (candidates already sorted) -----
__global__ void k_out(const unsigned* __restrict__ keepw, const float* __restrict__ rec,
                      float* __restrict__ out) {
  int t = threadIdx.x;
  for (int i = t; i < 600; i += blockDim.x) out[i] = (i >= 500) ? -1.0f : 0.0f;
  __syncthreads();
  if (t == 0) {
    int rank = 0;
    for (int i = 0; i < MSEL && rank < NTOP; ++i) {
      if ((keepw[i >> 5] >> (i & 31)) & 1u) {
        const float* rp = rec + (size_t)i * RECF;
        out[rank * 4 + 0] = rp[0];
        out[rank * 4 + 1] = rp[1];
        out[rank * 4 + 2] = rp[2];
        out[rank * 4 + 3] = rp[3];
        out[400 + rank] = rp[6];
        out[500 + rank] = rp[5];
        ++rank;
      }
    }
  }
}

extern "C" void kernel_launch(void* const* d_in, const int* in_sizes, int n_in,
                              void* d_out, int out_size, void* d_ws, size_t ws_size,
                              hipStream_t stream) {
  (void)in_sizes; (void)n_in; (void)out_size; (void)ws_size;
  const float* boxes  = (const float*)d_in[0];
  const float* scores = (const float*)d_in[1];
  char* ws = (char*)d_ws;
  unsigned*           hist   = (unsigned*)(ws + O_HIST);
  unsigned*           cnt    = (unsigned*)(ws + O_CNT);
  float*              thr    = (float*)(ws + O_THR);
  unsigned*           validw = (unsigned*)(ws + O_VALID);
  unsigned*           keepw  = (unsigned*)(ws + O_KEEP);
  unsigned long long* keys   = (unsigned long long*)(ws + O_KEYS);
  float*              rec    = (float*)(ws + O_REC);
  unsigned*           sup    = (unsigned*)(ws + O_SUP);

  k_init<<<(NB + 255) / 256, 256, 0, stream>>>((unsigned*)ws);
  k_hist<<<1024, 256, 0, stream>>>(scores, hist);
  k_thresh<<<1, 32, 0, stream>>>(hist, thr);
  k_collect<<<1024, 256, 0, stream>>>(scores, thr, cnt, keys);
  k_sortgather<<<1, 1024, 0, stream>>>(boxes, cnt, keys, rec, validw);
  k_sup<<<(MSEL / 16) * (MSEL / 32), 32, 0, stream>>>(rec, sup);
  k_nms<<<1, 64, 0, stream>>>(sup, validw, keepw);
  k_out<<<1, 128, 0, stream>>>(keepw, rec, (float*)d_out);
}